// CELossWithSVLS_VE_26723286516275
// MI455X (gfx1250) — compile-verified
//
#include <hip/hip_runtime.h>
#include <hip/hip_bf16.h>

// Problem constants (fixed by reference setup_inputs): B=2, C=Cin=4, D=H=W=64.
#define CSTRIDE (64 * 64 * 64)      // per-channel / per-batch-label stride
#define NVOX    (2 * 64 * 64 * 64)  // total voxels

typedef float v2f __attribute__((ext_vector_type(2)));
typedef float v8f __attribute__((ext_vector_type(8)));

// CDNA5 async global->LDS path (gfx1250): gate on builtin availability so the
// compile can never regress if the toolchain lacks these builtins.
#if defined(__HIP_DEVICE_COMPILE__) && defined(__gfx1250__) &&                 \
    __has_builtin(__builtin_amdgcn_global_load_async_to_lds_b32) &&            \
    __has_builtin(__builtin_amdgcn_s_wait_asynccnt)
#define USE_ASYNC_LDS 1
#else
#define USE_ASYNC_LDS 0
#endif

#if USE_ASYNC_LDS
#define GLOBAL_AS __attribute__((address_space(1)))
#define LDS_AS    __attribute__((address_space(3)))
// Builtin signature (from clang diagnostic): (AS1 int*, AS3 int*, Imm, Imm).
// Cast chain: drop const -> generic void* -> addrspacecast to target AS void*
// -> retype to int* in the same AS. (Never convert AS via integers: generic
// LDS pointers are aperture-based, AS3 pointers are LDS offsets.)
__device__ __forceinline__ void async_g2l_b32(const void* g, void* l) {
    __builtin_amdgcn_global_load_async_to_lds_b32(
        (GLOBAL_AS int*)(GLOBAL_AS void*)(void*)g,
        (LDS_AS int*)(LDS_AS void*)l,
        /*offset=*/0, /*cpol=*/0);
}
#endif

// ---------------------------------------------------------------------------
// Pass 1: one block per 8x8x8 tile. Halo (10x10x10) of 4-channel image +
// labels staged in LDS (async on CDNA5). Per voxel: single pass over the 27
// offsets accumulating
//   G = sum g[o],  P = sum g[o]*x[label(nbr(o))]
// loss_vox = logsumexp(x) - invZ*((P-xc)/G + (1 - 1/G + EPS)*xc)
// Block sum written to partial[blockIdx.x].
// ---------------------------------------------------------------------------
__global__ __launch_bounds__(256) void svls_ce_pass1(
    const float* __restrict__ inp,   // [2,4,64,64,64] logits
    const int*   __restrict__ lab,   // [2,64,64,64]
    const float* __restrict__ img,   // [2,4,64,64,64]
    float* __restrict__ partial)     // [gridDim.x]
{
    __shared__ float simg[4 * 1000];  // 4 channels x 10^3 halo
    __shared__ int   slab[1000];
    __shared__ float wsum[8];

    const int tid = threadIdx.x;
    const int blk = blockIdx.x;
    const int b   = blk >> 9;           // 512 tiles per batch
    const int rem = blk & 511;
    const int tz0 = (rem >> 6) << 3;
    const int ty0 = ((rem >> 3) & 7) << 3;
    const int tx0 = (rem & 7) << 3;

    const float* __restrict__ imgb = img + (size_t)b * 4 * CSTRIDE;
    const int*   __restrict__ labb = lab + (size_t)b * CSTRIDE;
    const float* __restrict__ inpb = inp + (size_t)b * 4 * CSTRIDE;

    // --- halo fill with replicate (clamped) padding ---
    for (int l = tid; l < 1000; l += 256) {
        int lz = l / 100;
        int r  = l - lz * 100;
        int ly = r / 10;
        int lx = r - ly * 10;
        int gz = min(max(tz0 - 1 + lz, 0), 63);
        int gy = min(max(ty0 - 1 + ly, 0), 63);
        int gx = min(max(tx0 - 1 + lx, 0), 63);
        int gidx = (gz << 12) + (gy << 6) + gx;
#if USE_ASYNC_LDS
        // Per-lane async copies: global -> LDS without VGPR staging; the
        // clamped-replicate gather cannot be expressed as a TDM descriptor,
        // so per-lane GLOBAL_LOAD_ASYNC_TO_LDS_B32 is the right CDNA5 tool.
        async_g2l_b32(imgb + 0 * CSTRIDE + gidx, &simg[0 * 1000 + l]);
        async_g2l_b32(imgb + 1 * CSTRIDE + gidx, &simg[1 * 1000 + l]);
        async_g2l_b32(imgb + 2 * CSTRIDE + gidx, &simg[2 * 1000 + l]);
        async_g2l_b32(imgb + 3 * CSTRIDE + gidx, &simg[3 * 1000 + l]);
        async_g2l_b32(labb + gidx, &slab[l]);
#else
        simg[0 * 1000 + l] = imgb[0 * CSTRIDE + gidx];
        simg[1 * 1000 + l] = imgb[1 * CSTRIDE + gidx];
        simg[2 * 1000 + l] = imgb[2 * CSTRIDE + gidx];
        simg[3 * 1000 + l] = imgb[3 * CSTRIDE + gidx];
        slab[l] = labb[gidx];
#endif
    }

    // Prefetch this thread's logits (global_prefetch_b8); overlaps with the
    // in-flight async LDS fill.
    {
        int v0 = tid, v1 = tid + 256;
        int g0 = ((tz0 + (v0 >> 6)) << 12) + ((ty0 + ((v0 >> 3) & 7)) << 6) + (tx0 + (v0 & 7));
        int g1 = ((tz0 + (v1 >> 6)) << 12) + ((ty0 + ((v1 >> 3) & 7)) << 6) + (tx0 + (v1 & 7));
        __builtin_prefetch(inpb + g0, 0, 1);
        __builtin_prefetch(inpb + CSTRIDE + g0, 0, 1);
        __builtin_prefetch(inpb + 2 * CSTRIDE + g0, 0, 1);
        __builtin_prefetch(inpb + 3 * CSTRIDE + g0, 0, 1);
        __builtin_prefetch(inpb + g1, 0, 1);
        __builtin_prefetch(inpb + CSTRIDE + g1, 0, 1);
        __builtin_prefetch(inpb + 2 * CSTRIDE + g1, 0, 1);
        __builtin_prefetch(inpb + 3 * CSTRIDE + g1, 0, 1);
    }
#if USE_ASYNC_LDS
    __builtin_amdgcn_s_wait_asynccnt(0);  // this wave's LDS writes have landed
#endif
    __syncthreads();                      // cross-wave visibility

    // exp(-0.5*d2) table, d2 in {0,1,2,3}; indexed by compile-time constant.
    const float DWt[4] = {1.0f, 0.60653065971f, 0.36787944117f, 0.22313016014f};
    const float EPSf = 1e-16f;

    float loss = 0.0f;
#pragma unroll
    for (int i = 0; i < 2; ++i) {
        int v  = tid + i * 256;
        int lz = v >> 6;
        int ly = (v >> 3) & 7;
        int lx = v & 7;
        int cidx = (lz + 1) * 100 + (ly + 1) * 10 + (lx + 1);
        float c0 = simg[cidx];
        float c1 = simg[1000 + cidx];
        float c2 = simg[2000 + cidx];
        float c3 = simg[3000 + cidx];

        int gidx = ((tz0 + lz) << 12) + ((ty0 + ly) << 6) + (tx0 + lx);
        float x0 = inpb[gidx];
        float x1 = inpb[CSTRIDE + gidx];
        float x2 = inpb[2 * CSTRIDE + gidx];
        float x3 = inpb[3 * CSTRIDE + gidx];

        float G = 0.0f, P = 0.0f;
#pragma unroll
        for (int o = 0; o < 27; ++o) {
            const int dz = o / 9, dy = (o / 3) % 3, dx = o % 3;
            const int d2 = (dz - 1) * (dz - 1) + (dy - 1) * (dy - 1) + (dx - 1) * (dx - 1);
            int idx = (lz + dz) * 100 + (ly + dy) * 10 + (lx + dx);
            float diff = simg[idx] - c0;
            diff = fmaxf(diff, simg[1000 + idx] - c1);
            diff = fmaxf(diff, simg[2000 + idx] - c2);
            diff = fmaxf(diff, simg[3000 + idx] - c3);
            float g = __expf(-0.5f * diff * diff) * DWt[d2];
            int lb = slab[idx];
            float xv = (lb == 0) ? x0 : (lb == 1) ? x1 : (lb == 2) ? x2 : x3;
            G += g;
            P = fmaf(g, xv, P);
        }
        int   lbc = slab[cidx];
        float xc  = (lbc == 0) ? x0 : (lbc == 1) ? x1 : (lbc == 2) ? x2 : x3;
        float wc   = 1.0f / G;                              // g[center] == 1
        float invZ = 1.0f / (2.0f * (1.0f - wc) + EPSf);
        float S    = invZ * ((P - xc) * wc + (1.0f - wc + EPSf) * xc);
        // logsumexp over 4 classes
        float m   = fmaxf(fmaxf(x0, x1), fmaxf(x2, x3));
        float lse = m + __logf(__expf(x0 - m) + __expf(x1 - m) +
                               __expf(x2 - m) + __expf(x3 - m));
        loss += lse - S;
    }

    // wave32 reduction, then cross-wave via LDS
    for (int off = 16; off; off >>= 1) loss += __shfl_xor(loss, off, 32);
    if ((tid & 31) == 0) wsum[tid >> 5] = loss;
    __syncthreads();
    if (tid == 0) {
        float t = 0.0f;
#pragma unroll
        for (int w = 0; w < 8; ++w) t += wsum[w];
        partial[blk] = t;
    }
}

// ---------------------------------------------------------------------------
// Pass 2: one wave (32 lanes, EXEC all-ones). Strided accumulate, then a
// v_wmma_f32_16x16x4_f32 with B = ones acts as a 64-wide adder tree:
// A columns 0 (lanes 0-15, vgpr0) and 2 (lanes 16-31, vgpr0) hold the 32 lane
// partials; D[m,0] = s[m] + s[m+16]; lane0 holds D[0..7,0] in d0..d7 and
// lane16 holds D[8..15,0], so the grand total = t(lane0) + t(lane16).
// ---------------------------------------------------------------------------
__global__ __launch_bounds__(32) void svls_ce_pass2(
    const float* __restrict__ partial, int n, float* __restrict__ out, float scale)
{
    int lane = threadIdx.x;
    float s = 0.0f;
    for (int i = lane; i < n; i += 32) s += partial[i];

#if defined(__HIP_DEVICE_COMPILE__) && defined(__gfx1250__) && \
    __has_builtin(__builtin_amdgcn_wmma_f32_16x16x4_f32)
    v2f a;  a.x = s;    a.y = 0.0f;
    v2f bb; bb.x = 1.0f; bb.y = 1.0f;
    v8f c = {};
    c = __builtin_amdgcn_wmma_f32_16x16x4_f32(
            /*neg_a=*/false, a, /*neg_b=*/false, bb,
            /*c_mod=*/(short)0, c, /*reuse_a=*/false, /*reuse_b=*/false);
    float t = c[0] + c[1] + c[2] + c[3] + c[4] + c[5] + c[6] + c[7];
    float tot = __shfl(t, 0, 32) + __shfl(t, 16, 32);
#else
    float tot = s;
    for (int off = 16; off; off >>= 1) tot += __shfl_xor(tot, off, 32);
#endif
    if (lane == 0) out[0] = tot * scale;
}

// ---------------------------------------------------------------------------
extern "C" void kernel_launch(void* const* d_in, const int* in_sizes, int n_in,
                              void* d_out, int out_size, void* d_ws, size_t ws_size,
                              hipStream_t stream)
{
    const float* inp = (const float*)d_in[0];  // [2,4,64,64,64] float32
    const int*   lab = (const int*)d_in[1];    // [2,64,64,64]   int
    const float* img = (const float*)d_in[2];  // [2,4,64,64,64] float32
    float* out     = (float*)d_out;
    float* partial = (float*)d_ws;             // nblocks floats of scratch

    const int nblocks = 2 * 8 * 8 * 8;         // B * (64/8)^3 = 1024
    svls_ce_pass1<<<nblocks, 256, 0, stream>>>(inp, lab, img, partial);
    svls_ce_pass2<<<1, 32, 0, stream>>>(partial, nblocks, out, 1.0f / (float)NVOX);
}